// DiscreteGNN2_4157528343206
// MI455X (gfx1250) — compile-verified
//
#include <hip/hip_runtime.h>

typedef float v2f __attribute__((ext_vector_type(2)));
typedef float v8f __attribute__((ext_vector_type(8)));

#define D      128
#define D2     256
#define MT     64
#define LDA    132   // LDS row stride for A tile (pads banks: 4*m pattern)
#define LDH    260   // LDS row stride for hid tile
#define LNUM   5
#define BN_EPS 1e-5f

// ---------------------------------------------------------------- utilities
__global__ void k_zero(float* __restrict__ p, long n) {
    long i = (long)blockIdx.x * blockDim.x + threadIdx.x;
    long stride = (long)gridDim.x * blockDim.x;
    for (; i < n; i += stride) p[i] = 0.f;
}

// ------------------------------------------------- input node embedding (h0)
__global__ void k_embed(const int* __restrict__ x,
                        const float* __restrict__ emb1,
                        const float* __restrict__ emb2,
                        float* __restrict__ h, int N, int Npad)
{
    int node = blockIdx.x * 8 + (threadIdx.x >> 5);
    int lane = threadIdx.x & 31;
    if (node >= Npad) return;
    float4* out = (float4*)(h + (size_t)node * D);
    if (node >= N) { out[lane] = make_float4(0.f, 0.f, 0.f, 0.f); return; }
    int i1 = x[node * 2], i2 = x[node * 2 + 1];
    float4 a = ((const float4*)(emb1 + (size_t)i1 * D))[lane];
    float4 b = ((const float4*)(emb2 + (size_t)i2 * D))[lane];
    out[lane] = make_float4(a.x + b.x, a.y + b.y, a.z + b.z, a.w + b.w);
}

// ----------------------- message + scatter-add (BN+ReLU of prev layer fused)
__global__ void k_message(const float* __restrict__ h, float* __restrict__ agg,
                          const int* __restrict__ src, const int* __restrict__ dst,
                          const int* __restrict__ ea,
                          const float* __restrict__ e1, const float* __restrict__ e2,
                          const float* __restrict__ scale, const float* __restrict__ shift,
                          int use_bn, int E, int N)
{
    int item = blockIdx.x * 8 + (threadIdx.x >> 5);   // one wave per edge/self-loop
    int lane = threadIdx.x & 31;                      // 4 channels per lane
    if (item >= E + N) return;
    int s, d, a0, a1;
    if (item < E) { s = src[item]; d = dst[item]; a0 = ea[item * 2]; a1 = ea[item * 2 + 1]; }
    else          { s = item - E;  d = s;          a0 = 4;           a1 = 0; }
    float4 hv = ((const float4*)(h + (size_t)s * D))[lane];
    if (use_bn) {  // intermediate layers: h_in = relu(bn(h_prev))
        float4 sc = ((const float4*)scale)[lane];
        float4 sh = ((const float4*)shift)[lane];
        hv.x = fmaxf(hv.x * sc.x + sh.x, 0.f);
        hv.y = fmaxf(hv.y * sc.y + sh.y, 0.f);
        hv.z = fmaxf(hv.z * sc.z + sh.z, 0.f);
        hv.w = fmaxf(hv.w * sc.w + sh.w, 0.f);
    }
    float4 ev1 = ((const float4*)(e1 + (size_t)a0 * D))[lane];
    float4 ev2 = ((const float4*)(e2 + (size_t)a1 * D))[lane];
    float* ap = agg + (size_t)d * D + lane * 4;
    atomicAdd(ap + 0, hv.x + ev1.x + ev2.x);
    atomicAdd(ap + 1, hv.y + ev1.y + ev2.y);
    atomicAdd(ap + 2, hv.z + ev1.z + ev2.z);
    atomicAdd(ap + 3, hv.w + ev1.w + ev2.w);
}

// -------- fused 2-layer MLP via V_WMMA_F32_16X16X4_F32, in-place, + BN stats
__global__ __launch_bounds__(256)
void k_mlp(float* __restrict__ h,           // in: agg [Npad][128]; out: pre-BN h
           const float* __restrict__ W1, const float* __restrict__ b1,
           const float* __restrict__ W2, const float* __restrict__ b2,
           float* __restrict__ stats, int N)
{
    __shared__ float aT[MT * LDA];
    __shared__ float hT[MT * LDH];
    __shared__ float ssum[D], ssq[D];

    const int tid  = threadIdx.x;
    const int wave = tid >> 5, lane = tid & 31;
    const int rowBase = blockIdx.x * MT;
    const int nl   = lane & 15;
    const int half = lane >> 4;       // 0: lanes 0-15, 1: lanes 16-31
    const int klo  = half * 2;        // A/B fragment K offset within 4-step

    // stage 64x128 agg tile (coalesced float4)
    for (int i = tid; i < MT * (D / 4); i += 256) {
        int r = i >> 5, c4 = i & 31;
        *(float4*)(aT + r * LDA + c4 * 4) =
            *(const float4*)(h + (size_t)(rowBase + r) * D + c4 * 4);
    }
    if (tid < D) { ssum[tid] = 0.f; ssq[tid] = 0.f; }
    __syncthreads();

    const int rgrp = wave & 3;            // 4 row-groups of 16
    const int mrow = rgrp * 16 + nl;      // A fragment row for this lane

    // ---- GEMM1: hid[64x256] = relu(agg @ W1 + b1); wave covers 8 col tiles
    const int cbase1 = (wave >> 2) * 128;
    #pragma unroll
    for (int jt = 0; jt < 8; ++jt) {
        const int n0 = cbase1 + jt * 16;
        v8f acc = {0.f, 0.f, 0.f, 0.f, 0.f, 0.f, 0.f, 0.f};
        #pragma unroll 4
        for (int k0 = 0; k0 < D; k0 += 4) {
            v2f a, b;
            a.x = aT[mrow * LDA + k0 + klo];
            a.y = aT[mrow * LDA + k0 + klo + 1];
            b.x = W1[(size_t)(k0 + klo) * D2 + n0 + nl];
            b.y = W1[(size_t)(k0 + klo + 1) * D2 + n0 + nl];
            acc = __builtin_amdgcn_wmma_f32_16x16x4_f32(false, a, false, b,
                                                        (short)0, acc, false, false);
        }
        const int  n    = n0 + nl;
        const float bias = b1[n];
        #pragma unroll
        for (int v = 0; v < 8; ++v) {
            int m = rgrp * 16 + half * 8 + v;   // C layout: vgpr v -> M=v (+8 hi half)
            hT[m * LDH + n] = fmaxf(acc[v] + bias, 0.f);
        }
    }
    __syncthreads();

    // ---- GEMM2: out[64x128] = hid @ W2 + b2; wave covers 4 col tiles
    const int cbase2 = (wave >> 2) * 64;
    #pragma unroll
    for (int jt = 0; jt < 4; ++jt) {
        const int n0 = cbase2 + jt * 16;
        v8f acc = {0.f, 0.f, 0.f, 0.f, 0.f, 0.f, 0.f, 0.f};
        #pragma unroll 4
        for (int k0 = 0; k0 < D2; k0 += 4) {
            v2f a, b;
            a.x = hT[mrow * LDH + k0 + klo];
            a.y = hT[mrow * LDH + k0 + klo + 1];
            b.x = W2[(size_t)(k0 + klo) * D + n0 + nl];
            b.y = W2[(size_t)(k0 + klo + 1) * D + n0 + nl];
            acc = __builtin_amdgcn_wmma_f32_16x16x4_f32(false, a, false, b,
                                                        (short)0, acc, false, false);
        }
        const int  n    = n0 + nl;
        const float bias = b2[n];
        float lsum = 0.f, lsq = 0.f;
        #pragma unroll
        for (int v = 0; v < 8; ++v) {
            int m = rgrp * 16 + half * 8 + v;
            float val = acc[v] + bias;
            int grow = rowBase + m;
            h[(size_t)grow * D + n] = val;              // in-place overwrite (tile already in LDS)
            if (grow < N) { lsum += val; lsq += val * val; }  // mask padding out of BN stats
        }
        atomicAdd(&ssum[n], lsum);   // ds_add_f32
        atomicAdd(&ssq[n],  lsq);
    }
    __syncthreads();
    if (tid < D) {
        atomicAdd(&stats[tid],     ssum[tid]);
        atomicAdd(&stats[D + tid], ssq[tid]);
    }
}

// ------------------------- BN batch stats -> per-channel affine scale/shift
__global__ void k_bnparams(const float* __restrict__ stats,
                           const float* __restrict__ gamma, const float* __restrict__ beta,
                           float* __restrict__ scale, float* __restrict__ shift, int N)
{
    int c = threadIdx.x;
    float mu  = stats[c] / (float)N;
    float var = stats[D + c] / (float)N - mu * mu;   // biased variance (jnp.var)
    float inv = rsqrtf(var + BN_EPS);
    float sc  = gamma[c] * inv;
    scale[c] = sc;
    shift[c] = beta[c] - mu * sc;
}

// -------- final: apply last BN (no relu), emit node repr + pooled sums
__global__ void k_final(const float* __restrict__ h,
                        const float* __restrict__ scale, const float* __restrict__ shift,
                        const int* __restrict__ batch,
                        float* __restrict__ node_out, float* __restrict__ pool, int N)
{
    int node = blockIdx.x * 8 + (threadIdx.x >> 5);
    int lane = threadIdx.x & 31;
    if (node >= N) return;
    float4 hv = ((const float4*)(h + (size_t)node * D))[lane];
    float4 sc = ((const float4*)scale)[lane];
    float4 sh = ((const float4*)shift)[lane];
    float4 v  = make_float4(hv.x * sc.x + sh.x, hv.y * sc.y + sh.y,
                            hv.z * sc.z + sh.z, hv.w * sc.w + sh.w);
    ((float4*)(node_out + (size_t)node * D))[lane] = v;
    int g = batch[node];
    float* pp = pool + (size_t)g * D + lane * 4;
    atomicAdd(pp + 0, v.x); atomicAdd(pp + 1, v.y);
    atomicAdd(pp + 2, v.z); atomicAdd(pp + 3, v.w);
}

__global__ void k_count(const int* __restrict__ batch, float* __restrict__ cnt, int N) {
    int i = blockIdx.x * blockDim.x + threadIdx.x;
    if (i < N) atomicAdd(&cnt[batch[i]], 1.f);
}

__global__ void k_pooldiv(const float* __restrict__ pool, const float* __restrict__ cnt,
                          float* __restrict__ out, int G) {
    int i = blockIdx.x * blockDim.x + threadIdx.x;
    if (i < G * D) out[i] = pool[i] / fmaxf(cnt[i >> 7], 1.f);
}

// --------------------------------------------------------------------------
extern "C" void kernel_launch(void* const* d_in, const int* in_sizes, int n_in,
                              void* d_out, int out_size, void* d_ws, size_t ws_size,
                              hipStream_t stream)
{
    const int*   x     = (const int*)  d_in[0];
    const int*   eidx  = (const int*)  d_in[1];
    const int*   eattr = (const int*)  d_in[2];
    const int*   batch = (const int*)  d_in[3];
    const float* xe1   = (const float*)d_in[4];
    const float* xe2   = (const float*)d_in[5];
    const float* ee1   = (const float*)d_in[6];   // [L][7][128]
    const float* ee2   = (const float*)d_in[7];   // [L][3][128]
    const float* W1    = (const float*)d_in[8];   // [L][128][256]
    const float* b1    = (const float*)d_in[9];   // [L][256]
    const float* W2    = (const float*)d_in[10];  // [L][256][128]
    const float* b2    = (const float*)d_in[11];  // [L][128]
    const float* gamma = (const float*)d_in[12];  // [L][128]
    const float* beta  = (const float*)d_in[13];  // [L][128]
    float* out = (float*)d_out;

    const int N    = in_sizes[0] / 2;
    const int E    = in_sizes[1] / 2;
    const int G    = out_size / D - N;            // out = [G*D] ++ [N*D]
    const int Npad = ((N + MT - 1) / MT) * MT;

    float* ws    = (float*)d_ws;
    float* bufA  = ws;
    float* bufB  = bufA  + (size_t)Npad * D;
    float* stats = bufB  + (size_t)Npad * D;      // [2*D]
    float* scale = stats + 2 * D;                 // [D]
    float* shift = scale + D;                     // [D]
    float* pool  = shift + D;                     // [G*D]
    float* cnt   = pool  + (size_t)G * D;         // [G]

    const int* srcp = eidx;
    const int* dstp = eidx + E;

    dim3 blk(256);
    k_embed<<<(Npad + 7) / 8, blk, 0, stream>>>(x, xe1, xe2, bufA, N, Npad);

    float* hcur = bufA; float* hnext = bufB;
    const int items = E + N;
    for (int l = 0; l < LNUM; ++l) {
        k_zero<<<2048, blk, 0, stream>>>(hnext, (long)Npad * D);
        k_message<<<(items + 7) / 8, blk, 0, stream>>>(hcur, hnext, srcp, dstp, eattr,
                ee1 + (size_t)l * 7 * D, ee2 + (size_t)l * 3 * D,
                scale, shift, (l > 0) ? 1 : 0, E, N);
        k_zero<<<1, blk, 0, stream>>>(stats, 2 * D);
        k_mlp<<<Npad / MT, blk, 0, stream>>>(hnext,
                W1 + (size_t)l * D * D2, b1 + (size_t)l * D2,
                W2 + (size_t)l * D2 * D, b2 + (size_t)l * D, stats, N);
        k_bnparams<<<1, D, 0, stream>>>(stats, gamma + (size_t)l * D,
                                        beta + (size_t)l * D, scale, shift, N);
        float* t = hcur; hcur = hnext; hnext = t;
    }

    k_zero<<<2048, blk, 0, stream>>>(pool, (long)G * D + G);  // pool + cnt contiguous
    k_final<<<(N + 7) / 8, blk, 0, stream>>>(hcur, scale, shift, batch,
                                             out + (size_t)G * D, pool, N);
    k_count<<<(N + 255) / 256, blk, 0, stream>>>(batch, cnt, N);
    k_pooldiv<<<(G * D + 255) / 256, blk, 0, stream>>>(pool, cnt, out, G);
}